// HP_7481833029981
// MI455X (gfx1250) — compile-verified
//
#include <hip/hip_runtime.h>
#include <stdint.h>

// Problem constants (from reference): B=8, C=32, W=64
#define WDIM   64
#define HDIM   32
#define NVOX   (WDIM * WDIM * WDIM)   // 262144 elements per (b,c) slice (1 MB f32)
#define NPOOL  (HDIM * HDIM * HDIM)   // 32768 pooled elements per slice
#define NBC    256                    // B*C
#define TPB    256                    // 8 wave32s
#define JBLK   (NPOOL / TPB)          // 128 j-blocks
#define BCG    8                      // (b,c) slices per block

typedef unsigned int u32;
typedef u32 v4u __attribute__((ext_vector_type(4)));
typedef int  v8i __attribute__((ext_vector_type(8)));
typedef int  v4i __attribute__((ext_vector_type(4)));

// ---------------------------------------------------------------------------
// Main kernel (placed FIRST so the disasm snippet shows the TDM sequence):
// TDM stages the block's 8 KB index tile (2D tile: 256 wide x 8 rows, row
// stride 32768 ints) into LDS; then each thread max-reduces its 8 gathered
// values for 8 consecutive (b,c) slices, prefetching the next slice's stripe
// into L2 while gathering the current one. bc-loop unrolled 2x -> 16
// independent gathers in flight per thread.
// ---------------------------------------------------------------------------
__global__ void __launch_bounds__(TPB)
pool_tdm(const float* __restrict__ x,
         const int*  __restrict__ fused,
         float*      __restrict__ out) {
    __shared__ __align__(16) int sIdx[8 * TPB];       // 8 KB, SoA [k][256]
    const int tid = threadIdx.x;
    const int jb  = blockIdx.x;                       // 0..JBLK-1
    const int j   = jb * TPB + tid;                   // pooled index
    const int bc0 = blockIdx.y * BCG;

    // Warm L2 with this block's stripe of the FIRST slice while TDM runs.
    if (tid < 64)
        __builtin_prefetch(x + (size_t)bc0 * NVOX + jb * 2048 + tid * 32, 0, 1);

    if (tid < 32) {   // wave 0 only issues the tensor DMA (EXEC ignored by TDM)
        const unsigned long long ga =
            (unsigned long long)(uintptr_t)(fused + (size_t)jb * TPB);
        const u32 lds = (u32)(uintptr_t)&sIdx[0];

        // D# group 0: count=1, lds_addr, global_addr[56:0], type=2
        v4u g0;
        g0[0] = 1u;
        g0[1] = lds;
        g0[2] = (u32)(ga & 0xFFFFFFFFull);
        g0[3] = ((u32)(ga >> 32) & 0x01FFFFFFu) | 0x80000000u;

        // D# group 1: data_size=4B; tensor 32768x8, tile 256x8, stride 32768
        v8i g1;
        g1[0] = 0x00020000;            // workgroup_mask=0, data_size=2 (4B)
        g1[1] = (int)0x80000000u;      // tensor_dim0 = 32768 (bits 79:48)
        g1[2] = 0x00080000;            // tensor_dim0 hi=0, tensor_dim1 = 8
        g1[3] = 0x01000000;            // tensor_dim1 hi=0, tile_dim0 = 256
        g1[4] = 0x00000008;            // tile_dim1 = 8, tile_dim2 = 0
        g1[5] = 0x00008000;            // tensor_dim0_stride = 32768 (lo)
        g1[6] = 0;                     // stride hi = 0, dim1_stride lo = 0
        g1[7] = 0;

        v4i gz = {0, 0, 0, 0};         // groups 2/3 unused (2D tensor)
#if __clang_major__ >= 23
        v8i gz8 = {0, 0, 0, 0, 0, 0, 0, 0};
        __builtin_amdgcn_tensor_load_to_lds(g0, g1, gz, gz, gz8, 0);
#else
        __builtin_amdgcn_tensor_load_to_lds(g0, g1, gz, gz, 0);
#endif
        __builtin_amdgcn_s_wait_tensorcnt(0);
    }
    __syncthreads();

    // Indices are loop-invariant: compiler hoists these 8 LDS reads into
    // VGPRs and reuses them across all BCG slices.
    int idx[8];
#pragma unroll
    for (int k = 0; k < 8; ++k)
        idx[k] = sIdx[k * TPB + tid];

#pragma unroll 2
    for (int i = 0; i < BCG; ++i) {
        const int bc = bc0 + i;
        const float* xs = x + (size_t)bc * NVOX;

        // Prefetch the next slice's stripe into L2 (sequential -> full lines).
        if (i + 1 < BCG && tid < 64)
            __builtin_prefetch(xs + NVOX + jb * 2048 + tid * 32, 0, 1);

        // 8 independent L2 gathers, max-reduce.
        float m = xs[idx[0]];
#pragma unroll
        for (int k = 1; k < 8; ++k)
            m = fmaxf(m, xs[idx[k]]);

        out[(size_t)bc * NPOOL + j] = m;   // coalesced store
    }
}

// ---------------------------------------------------------------------------
// Kernel 1: fuse the two permutations + pooling-window indexing into one
// SoA table:  fused[k*NPOOL + j] = distance[voxel(coordinate[j], k)]
// 1 MB total, computed once, reused by all 256 (b,c) slices.
// ---------------------------------------------------------------------------
__global__ void __launch_bounds__(TPB)
build_fused_idx(const int* __restrict__ coordinate,
                const int* __restrict__ distance,
                int* __restrict__ fused) {
    const int j = blockIdx.x * TPB + threadIdx.x;     // 0..NPOOL-1
    const int p = coordinate[j];                      // pooled 3D linear index
    const int z = ((p >> 10) & 31) << 1;              // 2*z in 64^3 cube
    const int y = ((p >> 5)  & 31) << 1;              // 2*y
    const int x = ( p        & 31) << 1;              // 2*x
#pragma unroll
    for (int k = 0; k < 8; ++k) {
        // fields are disjoint: (z+dz)<64, (y+dy)<64, (x+dx)<64
        const int v = ((z + (k >> 2)) << 12) | ((y + ((k >> 1) & 1)) << 6)
                    | (x + (k & 1));
        fused[k * NPOOL + j] = distance[v];
    }
}

// ---------------------------------------------------------------------------
// Fallback (no workspace): recompute indices inline per (b,c).
// ---------------------------------------------------------------------------
__global__ void __launch_bounds__(TPB)
pool_direct(const float* __restrict__ x,
            const int*  __restrict__ distance,
            const int*  __restrict__ coordinate,
            float*      __restrict__ out) {
    const int tid = threadIdx.x;
    const int j   = blockIdx.x * TPB + tid;
    const int bc  = blockIdx.y;
    const float* xs = x + (size_t)bc * NVOX;

    if (tid < 64)
        __builtin_prefetch(xs + blockIdx.x * 2048 + tid * 32, 0, 1);

    const int p = coordinate[j];
    const int z = ((p >> 10) & 31) << 1;
    const int y = ((p >> 5)  & 31) << 1;
    const int xx = (p        & 31) << 1;

    float m = -3.402823466e38f;
#pragma unroll
    for (int k = 0; k < 8; ++k) {
        const int v = ((z + (k >> 2)) << 12) | ((y + ((k >> 1) & 1)) << 6)
                    | (xx + (k & 1));
        m = fmaxf(m, xs[distance[v]]);
    }
    out[(size_t)bc * NPOOL + j] = m;
}

// ---------------------------------------------------------------------------
extern "C" void kernel_launch(void* const* d_in, const int* in_sizes, int n_in,
                              void* d_out, int out_size, void* d_ws, size_t ws_size,
                              hipStream_t stream) {
    const float* x          = (const float*)d_in[0];
    const int*   distance   = (const int*)d_in[1];
    const int*   coordinate = (const int*)d_in[2];
    float*       out        = (float*)d_out;

    if (ws_size >= (size_t)8 * NPOOL * sizeof(int)) {
        int* fused = (int*)d_ws;
        build_fused_idx<<<JBLK, TPB, 0, stream>>>(coordinate, distance, fused);
        pool_tdm<<<dim3(JBLK, NBC / BCG), TPB, 0, stream>>>(x, fused, out);
    } else {
        pool_direct<<<dim3(JBLK, NBC), TPB, 0, stream>>>(x, distance, coordinate, out);
    }
}